// RandomJointFlip_43327630082182
// MI455X (gfx1250) — compile-verified
//
#include <hip/hip_runtime.h>

typedef __attribute__((ext_vector_type(2))) float v2f;
typedef __attribute__((ext_vector_type(4))) float f4;
typedef __attribute__((ext_vector_type(8))) float v8f;

#define NB 8
#define DD 96
#define HH 192
#define WW 192
#define VOL (NB*DD*HH*WW)        /* 28,311,552 floats per tensor   */
#define V4  (VOL/4)              /* 7,077,888 float4 per tensor    */
#define RW4 (WW/4)               /* 48 float4 per W-row            */

// ---------------------------------------------------------------------------
// Flip matrix element T[k][c] for mode: 0 = identity, 1 = dflip (Td), 2 = hflip (Th)
// Th: eye with [0][0]=-1, [0][3]=W-1=191 ; Td: eye with [2][2]=-1, [2][3]=D-1=95
// ---------------------------------------------------------------------------
__device__ __forceinline__ float Tval(int mode, int k, int c) {
    float v = (k == c) ? 1.0f : 0.0f;
    if (mode == 2) {                 // hflip
        if (k == 0 && c == 0) v = -1.0f;
        if (k == 0 && c == 3) v = 191.0f;
    } else if (mode == 1) {          // dflip
        if (k == 2 && c == 2) v = -1.0f;
        if (k == 2 && c == 3) v = 95.0f;
    }
    return v;
}

// ---------------------------------------------------------------------------
// Kernel 1: one wave32. Lanes 0..7 classify the view per batch and derive flip
// modes; then the whole wave computes a1o = affine1@T1 and a2o = affine2@T2
// with V_WMMA_F32_16X16X4_F32 (4 batches of 4x4 GEMMs block-diagonally packed
// per WMMA; 4 WMMA issues total). Modes are exported to d_ws for kernel 2.
// ---------------------------------------------------------------------------
__global__ __launch_bounds__(32)
void flags_affine_wmma(const float* __restrict__ aff1,
                       const float* __restrict__ aff2,
                       const int*   __restrict__ flip_mask,
                       float*       __restrict__ out,
                       int*         __restrict__ ws)
{
    __shared__ int smode[2][NB];
    const int lane = threadIdx.x;

    if (lane < NB) {
        const float VW[3][6] = {
            {-0.04579317872402952f, -0.07575516143899245f, -0.061679127638288714f,
              0.02846023147369374f,  0.9765938788126243f,   1.310693443330605f},
            {-1.7524712268232336f,   1.6816010000146304f,  -0.10457518199789638f,
              0.7600968544610555f,   0.3062302844162186f,   0.2499470378689073f},
            { 1.5898357864427095f,  -1.9611154042037333f,   0.22473070108297732f,
             -0.9235759306577322f,  -1.417853830070638f,   -1.459610511848043f}};
        const float VB[3] = {0.0009407975978420954f, 0.3678460758384997f, -1.1950944844551188f};

        const float* a = aff1 + lane * 16;
        float r0 = a[0], r1 = a[4], r2 = a[8];   // affine[:, :3, 0]
        float c0 = a[1], c1 = a[5], c2 = a[9];   // affine[:, :3, 1]
        float rn = sqrtf(r0*r0 + r1*r1 + r2*r2);
        float cn = sqrtf(c0*c0 + c1*c1 + c2*c2);
        float o[6] = {r0/rn, r1/rn, r2/rn, c0/cn, c1/cn, c2/cn};
        float s[3];
        for (int j = 0; j < 3; ++j) {
            float acc = VB[j];
            for (int k = 0; k < 6; ++k) acc += VW[j][k] * o[k];
            s[j] = acc;
        }
        int arg = 0;
        if (s[1] > s[0]) arg = 1;
        if (s[2] > s[arg]) arg = 2;
        bool is_sag = (arg == 1);
        bool fm = flip_mask[lane] != 0;
        int m1 = fm ? (is_sag ? 1 : 2) : 0;   // set 1: dflip if sagittal else hflip
        int m2 = fm ? (is_sag ? 2 : 1) : 0;   // set 2: swapped
        smode[0][lane] = m1;  smode[1][lane] = m2;
        ws[lane] = m1;  ws[NB + lane] = m2;
    }
    __syncthreads();

    // Per-lane packing indices (wave32, EXEC all ones -> WMMA legal).
    const int Nc  = (lane < 16) ? lane : lane - 16;  // column for B/D, row for A
    const int g   = Nc >> 2;                         // which 4x4 block (batch-in-group)
    const int rc  = Nc & 3;                          // row (A) / col (B,D) inside block
    const int klo = (lane < 16) ? 0 : 2;             // K pair held by this lane

    const float*     affs[2]  = {aff1, aff2};
    const long long  obase[2] = {2LL * VOL, 4LL * VOL + 128};

    for (int s2 = 0; s2 < 2; ++s2) {
        for (int half = 0; half < 2; ++half) {
            const int bb = half * 4 + g;             // absolute batch index
            const float* A = affs[s2] + bb * 16;
            v2f av;  av.x = A[rc * 4 + klo];  av.y = A[rc * 4 + klo + 1];
            const int md = smode[s2][bb];
            v2f bv;  bv.x = Tval(md, klo, rc);  bv.y = Tval(md, klo + 1, rc);
            v8f acc = {};
            v8f d = __builtin_amdgcn_wmma_f32_16x16x4_f32(
                        false, av, false, bv, (short)0, acc, false, false);
            // Extract the diagonal block: rows M = 4g + r (r = 0..3), column N = 4g+rc.
            // Lanes 0..15 hold M = 0..7 (own g<2, vb = 4g in {0,4});
            // lanes 16..31 hold M = 8..15 (own g>=2, vb = 4g-8 in {0,4}).
            const bool own = (lane < 16) ? (g < 2) : (g >= 2);
            const bool lo4 = (lane < 16) ? (g == 0) : (g == 2);  // vb == 0
            if (own) {
                float* ob = out + obase[s2] + bb * 16 + rc;
                if (lo4) {
                    ob[0]  = d[0];  ob[4]  = d[1];
                    ob[8]  = d[2];  ob[12] = d[3];
                } else {
                    ob[0]  = d[4];  ob[4]  = d[5];
                    ob[8]  = d[6];  ob[12] = d[7];
                }
            }
        }
    }
}

// ---------------------------------------------------------------------------
// Kernel 2: streaming flip/copy of the four volumes with 128-bit non-temporal
// accesses. grid.y selects tensor (0:x1 1:m1 2:x2 3:m2).
// ---------------------------------------------------------------------------
__global__ __launch_bounds__(256)
void flip_copy(const float* __restrict__ x1, const float* __restrict__ m1,
               const float* __restrict__ x2, const float* __restrict__ m2,
               float* __restrict__ out, const int* __restrict__ ws)
{
    const long long i = (long long)blockIdx.x * blockDim.x + threadIdx.x;
    if (i >= (long long)V4) return;
    const int t = blockIdx.y;

    const int  w4  = (int)(i % RW4);
    const long long row = i / RW4;          // n*D*H + d*H + h
    const int  h   = (int)(row % HH);
    const long long dn = row / HH;
    const int  d   = (int)(dn % DD);
    const int  n   = (int)(dn / DD);

    const float* srcs[4]  = {x1, m1, x2, m2};
    const long long ob[4] = {0LL, (long long)VOL, 2LL*VOL + 128, 3LL*VOL + 128};
    const int mode = (t < 2) ? ws[n] : ws[NB + n];

    const f4* src = (const f4*)srcs[t];
    f4*       dst = (f4*)(out + ob[t]);

    f4 v;
    if (mode == 1) {                         // D flip: remap row, keep W order
        long long sidx = (((long long)n * DD + (DD - 1 - d)) * HH + h) * RW4 + w4;
        v = __builtin_nontemporal_load(src + sidx);
    } else if (mode == 2) {                  // W flip: mirrored vec4, reverse lanes
        f4 s = __builtin_nontemporal_load(src + row * RW4 + (RW4 - 1 - w4));
        v.x = s.w; v.y = s.z; v.z = s.y; v.w = s.x;
    } else {                                 // identity
        v = __builtin_nontemporal_load(src + i);
    }
    __builtin_nontemporal_store(v, dst + i);
}

// ---------------------------------------------------------------------------
extern "C" void kernel_launch(void* const* d_in, const int* in_sizes, int n_in,
                              void* d_out, int out_size, void* d_ws, size_t ws_size,
                              hipStream_t stream)
{
    const float* x1 = (const float*)d_in[0];
    const float* m1 = (const float*)d_in[1];
    const float* a1 = (const float*)d_in[2];
    const float* x2 = (const float*)d_in[3];
    const float* m2 = (const float*)d_in[4];
    const float* a2 = (const float*)d_in[5];
    const int*   fm = (const int*)  d_in[6];
    float* out = (float*)d_out;
    int*   ws  = (int*)d_ws;

    flags_affine_wmma<<<1, 32, 0, stream>>>(a1, a2, fm, out, ws);

    dim3 grid((V4 + 255) / 256, 4, 1);
    flip_copy<<<grid, 256, 0, stream>>>(x1, m1, x2, m2, out, ws);
}